// PixelShuffle1d_52785148068480
// MI455X (gfx1250) — compile-verified
//
#include <hip/hip_runtime.h>

// PixelShuffle1d: x (B=8, C=256, L=16384) f32 -> y (B, C/4, L*4)
//   y[b, c', l*4 + j] = x[b, c'*4 + j, l]
// Flat: with bc = b*64 + c',  out[bc*4L + l*4 + j] = in[bc*4L + j*L + l]
//
// Pure bandwidth problem: 128 MiB in + 128 MiB out. Strategy:
//  - per thread: 4 x global_load_b128 (float4 from each of 4 input rows,
//    coalesced across the wave), 4x4 register transpose,
//    4 x global_store_b128 (64 contiguous bytes per thread).
//  - non-temporal hints both ways: 256 MiB streamed once > 192 MB L2.

#define BB 8
#define CC 256
#define LL 16384
#define UU 4

typedef __attribute__((ext_vector_type(4))) float f32x4;

__global__ __launch_bounds__(256)
void pixel_shuffle1d_kernel(const float* __restrict__ x, float* __restrict__ y) {
    const int LG = LL / 4;                       // 4096 l-groups per (b,c')
    const int tid = blockIdx.x * blockDim.x + threadIdx.x;

    const int lg = tid & (LG - 1);               // which group of 4 l's
    const int bc = tid >> 12;                    // (b*64 + c'), tid / 4096

    const int base    = bc * (4 * LL);           // shared flat base (elements)
    const int in_off  = base + lg * 4;           // + l0, rows are LL apart
    const int out_off = base + lg * 16;          // + l0*4

    // 4 coalesced 16B loads, one per interleaved input row (64 KiB apart)
    f32x4 r0 = __builtin_nontemporal_load((const f32x4*)(x + in_off         ));
    f32x4 r1 = __builtin_nontemporal_load((const f32x4*)(x + in_off + 1 * LL));
    f32x4 r2 = __builtin_nontemporal_load((const f32x4*)(x + in_off + 2 * LL));
    f32x4 r3 = __builtin_nontemporal_load((const f32x4*)(x + in_off + 3 * LL));

    // 4x4 register transpose: o_l = (r0[l], r1[l], r2[l], r3[l])
    f32x4 o0 = { r0.x, r1.x, r2.x, r3.x };
    f32x4 o1 = { r0.y, r1.y, r2.y, r3.y };
    f32x4 o2 = { r0.z, r1.z, r2.z, r3.z };
    f32x4 o3 = { r0.w, r1.w, r2.w, r3.w };

    // 64 contiguous output bytes per thread -> 2 KiB contiguous per wave
    __builtin_nontemporal_store(o0, (f32x4*)(y + out_off +  0));
    __builtin_nontemporal_store(o1, (f32x4*)(y + out_off +  4));
    __builtin_nontemporal_store(o2, (f32x4*)(y + out_off +  8));
    __builtin_nontemporal_store(o3, (f32x4*)(y + out_off + 12));
}

extern "C" void kernel_launch(void* const* d_in, const int* in_sizes, int n_in,
                              void* d_out, int out_size, void* d_ws, size_t ws_size,
                              hipStream_t stream) {
    (void)in_sizes; (void)n_in; (void)d_ws; (void)ws_size; (void)out_size;

    const float* x = (const float*)d_in[0];
    float*       y = (float*)d_out;

    // 512 (b,c') pairs * 4096 l-groups = 2,097,152 threads
    const int total_threads = (BB * (CC / UU)) * (LL / 4);
    const int block = 256;                        // 8 wave32 waves
    const int grid  = total_threads / block;      // 8192 blocks

    pixel_shuffle1d_kernel<<<grid, block, 0, stream>>>(x, y);
}